// ChamferLossAdapted_29927332118898
// MI455X (gfx1250) — compile-verified
//
#include <hip/hip_runtime.h>
#include <math.h>

// ---------------------------------------------------------------------------
// Chamfer loss for MI455X (gfx1250, wave32).
//
// dist(n,m) = ||p1||^2 + ||p2||^2 - 2 p1.p2 via one V_WMMA_F32_16X16X4_F32
// per 16x16 tile (augmentation):
//   A row  = (-2x, -2y, -2z, ||p1||^2)   (16x4 f32 A)
//   B col  = ( x,   y,   z,  1       )   (4x16 f32 B)
//   C      = ||p2[m]||^2 broadcast       (16x16 f32 C)
// => D = A*B + C = squared-distance tile, K=4 fully used.
//
// Two transposed single-writer sweeps, each with:
//   - 32 owned rows/cols per wave (2 tiles -> 2 WMMAs per operand load)
//   - scanned dimension split into 8 slices for 8192/4096-wave parallelism
// Partial mins are folded in the final reduction kernel. No atomics.
// ---------------------------------------------------------------------------

typedef __attribute__((ext_vector_type(2))) float v2f;
typedef __attribute__((ext_vector_type(8))) float v8f;

#define B_ 2
#define N_ 16384
#define M_ 8192
#define NT_ (N_ / 16)      // 1024 row tiles per batch
#define MT_ (M_ / 16)      // 512  col tiles per batch
#define SLICES_ 8
#define BN_ (B_ * N_)
#define BM_ (B_ * M_)

// ---------------- prep: build augmented operand arrays ----------------
__global__ __launch_bounds__(256) void chamfer_prep(
    const float* __restrict__ p1, const float* __restrict__ p2,
    float* __restrict__ a1aug,   // (B*N, 4) = (-2x,-2y,-2z, sq1)
    float* __restrict__ a2aug,   // (B*M, 4) = ( x,  y,  z,  1 )
    float* __restrict__ sq2)     // (B*M)
{
  int idx = blockIdx.x * blockDim.x + threadIdx.x;
  if (idx < BN_) {
    float x = p1[idx * 3 + 0], y = p1[idx * 3 + 1], z = p1[idx * 3 + 2];
    float4 v; v.x = -2.f * x; v.y = -2.f * y; v.z = -2.f * z;
    v.w = x * x + y * y + z * z;
    ((float4*)a1aug)[idx] = v;
  }
  if (idx < BM_) {
    float x = p2[idx * 3 + 0], y = p2[idx * 3 + 1], z = p2[idx * 3 + 2];
    float4 v; v.x = x; v.y = y; v.z = z; v.w = 1.0f;
    ((float4*)a2aug)[idx] = v;
    sq2[idx] = x * x + y * y + z * z;
  }
}

// ---------------- rowmin: min over an M-slice for 32 owned rows ----------------
// Wave owns 2 row tiles (A0,A1 fixed); scans MT_/SLICES_ = 64 col tiles.
// Writes md1p[slice][b][n] partial mins.
__global__ __launch_bounds__(256) void chamfer_rowmin(
    const float* __restrict__ a1aug, const float* __restrict__ a2aug,
    const float* __restrict__ sq2, float* __restrict__ md1p)
{
  const int lane  = threadIdx.x & 31;
  const int wave  = threadIdx.x >> 5;
  const int wid   = blockIdx.x * 8 + wave;     // 0 .. B*(N/32)*SLICES-1
  const int slice = wid & (SLICES_ - 1);
  const int rp    = wid >> 3;
  const int b     = rp / (N_ / 32);
  const int nb    = (rp % (N_ / 32)) * 32;
  const int hi    = lane >> 4;                 // 0: K0,K1  1: K2,K3
  const int lo    = lane & 15;

  // A 16x4 f32 layout: lane lo holds row; VGPR0=K(2*hi), VGPR1=K(2*hi+1)
  const float* a1b = a1aug + (size_t)b * N_ * 4;
  v2f A0 = *(const v2f*)(a1b + (size_t)(nb + lo) * 4 + hi * 2);
  v2f A1 = *(const v2f*)(a1b + (size_t)(nb + 16 + lo) * 4 + hi * 2);

  v8f rm0, rm1;
#pragma unroll
  for (int i = 0; i < 8; ++i) { rm0[i] = __builtin_inff(); rm1[i] = __builtin_inff(); }

  const float* a2base  = a2aug + (size_t)b * M_ * 4;
  const float* sq2base = sq2 + (size_t)b * M_;
  const int mt0 = slice * (MT_ / SLICES_);
  const int mt1 = mt0 + (MT_ / SLICES_);

  for (int mt = mt0; mt < mt1; ++mt) {
    const int m0 = mt * 16;
    v2f Bv   = *(const v2f*)(a2base + (size_t)(m0 + lo) * 4 + hi * 2);
    float cs = sq2base[m0 + lo];
    v8f C;
#pragma unroll
    for (int i = 0; i < 8; ++i) C[i] = cs;

    v8f d0 = __builtin_amdgcn_wmma_f32_16x16x4_f32(
        false, A0, false, Bv, (short)0, C, false, false);
    v8f d1 = __builtin_amdgcn_wmma_f32_16x16x4_f32(
        false, A1, false, Bv, (short)0, C, false, false);
#pragma unroll
    for (int i = 0; i < 8; ++i) {
      rm0[i] = fminf(rm0[i], d0[i]);
      rm1[i] = fminf(rm1[i], d1[i]);
    }
  }

  // D layout: element(i,j) -> vgpr i%8, lane (i/8)*16 + j.
  // Row-min = min over j = butterfly across the 16 lanes of each half-wave.
#pragma unroll
  for (int mask = 1; mask < 16; mask <<= 1) {
#pragma unroll
    for (int i = 0; i < 8; ++i) {
      rm0[i] = fminf(rm0[i], __shfl_xor(rm0[i], mask, 32));
      rm1[i] = fminf(rm1[i], __shfl_xor(rm1[i], mask, 32));
    }
  }
  if (lo == 0) {
    float* dst = md1p + (size_t)slice * BN_ + (size_t)b * N_;
    const int r0 = nb + hi * 8;
#pragma unroll
    for (int i = 0; i < 8; ++i) {
      dst[r0 + i]      = rm0[i];
      dst[r0 + 16 + i] = rm1[i];
    }
  }
}

// ---------------- colmin: min over an N-slice for 32 owned cols ----------------
// Wave owns 2 col tiles (B0/C0, B1/C1 fixed); scans NT_/SLICES_ = 128 row tiles.
// Writes md2p[slice][b][m] partial mins.
__global__ __launch_bounds__(256) void chamfer_colmin(
    const float* __restrict__ a1aug, const float* __restrict__ a2aug,
    const float* __restrict__ sq2, float* __restrict__ md2p)
{
  const int lane  = threadIdx.x & 31;
  const int wave  = threadIdx.x >> 5;
  const int wid   = blockIdx.x * 8 + wave;     // 0 .. B*(M/32)*SLICES-1
  const int slice = wid & (SLICES_ - 1);
  const int cp    = wid >> 3;
  const int b     = cp / (M_ / 32);
  const int mb    = (cp % (M_ / 32)) * 32;
  const int hi    = lane >> 4;
  const int lo    = lane & 15;

  const float* a2b = a2aug + (size_t)b * M_ * 4;
  v2f B0 = *(const v2f*)(a2b + (size_t)(mb + lo) * 4 + hi * 2);
  v2f B1 = *(const v2f*)(a2b + (size_t)(mb + 16 + lo) * 4 + hi * 2);
  float cs0 = sq2[(size_t)b * M_ + mb + lo];
  float cs1 = sq2[(size_t)b * M_ + mb + 16 + lo];
  v8f C0, C1;
#pragma unroll
  for (int i = 0; i < 8; ++i) { C0[i] = cs0; C1[i] = cs1; }

  float cm0 = __builtin_inff(), cm1 = __builtin_inff();
  const float* a1base = a1aug + (size_t)b * N_ * 4;
  const int nt0 = slice * (NT_ / SLICES_);
  const int nt1 = nt0 + (NT_ / SLICES_);

  for (int nt = nt0; nt < nt1; ++nt) {
    const int r0 = nt * 16;
    v2f A = *(const v2f*)(a1base + (size_t)(r0 + lo) * 4 + hi * 2);
    v8f d0 = __builtin_amdgcn_wmma_f32_16x16x4_f32(
        false, A, false, B0, (short)0, C0, false, false);
    v8f d1 = __builtin_amdgcn_wmma_f32_16x16x4_f32(
        false, A, false, B1, (short)0, C1, false, false);
    float t0 = d0[0], t1 = d1[0];
#pragma unroll
    for (int i = 1; i < 8; ++i) { t0 = fminf(t0, d0[i]); t1 = fminf(t1, d1[i]); }
    cm0 = fminf(cm0, t0);
    cm1 = fminf(cm1, t1);
  }
  // lane j holds rows {0..7}, lane j+16 rows {8..15}: merge half-waves.
  cm0 = fminf(cm0, __shfl_xor(cm0, 16, 32));
  cm1 = fminf(cm1, __shfl_xor(cm1, 16, 32));
  if (hi == 0) {
    float* dst = md2p + (size_t)slice * BM_ + (size_t)b * M_;
    dst[mb + lo]      = cm0;
    dst[mb + 16 + lo] = cm1;
  }
}

// ---------------- final: slice-min fold + softmax-weighted sum + col mean ----
__global__ __launch_bounds__(1024) void chamfer_final(
    const float* __restrict__ weights, const float* __restrict__ md1p,
    const float* __restrict__ md2p, float* __restrict__ out)
{
  __shared__ float sA[1024];
  __shared__ float sB[1024];
  const int tid = threadIdx.x;
  float total = 0.f;

  for (int b = 0; b < B_; ++b) {
    const float* w = weights + (size_t)b * N_;
    // 1) max over weights
    float lmax = -__builtin_inff();
    for (int i = tid; i < N_; i += 1024) lmax = fmaxf(lmax, w[i]);
    sA[tid] = lmax; __syncthreads();
    for (int s = 512; s > 0; s >>= 1) {
      if (tid < s) sA[tid] = fmaxf(sA[tid], sA[tid + s]);
      __syncthreads();
    }
    const float wmax = sA[0]; __syncthreads();
    // 2) softmax denom + weighted min_dist1 sum (slice-min folded here)
    float ls = 0.f, lt = 0.f;
    for (int i = tid; i < N_; i += 1024) {
      float e  = expf(w[i] - wmax);
      float m1 = md1p[(size_t)b * N_ + i];
#pragma unroll
      for (int s = 1; s < SLICES_; ++s)
        m1 = fminf(m1, md1p[(size_t)s * BN_ + (size_t)b * N_ + i]);
      ls += e;
      lt += e * m1;
    }
    sA[tid] = ls; sB[tid] = lt; __syncthreads();
    for (int s = 512; s > 0; s >>= 1) {
      if (tid < s) { sA[tid] += sA[tid + s]; sB[tid] += sB[tid + s]; }
      __syncthreads();
    }
    const float loss1 = sB[0] / sA[0]; __syncthreads();
    // 3) mean of min_dist2 (slice-min folded here)
    float lc = 0.f;
    for (int i = tid; i < M_; i += 1024) {
      float m2 = md2p[(size_t)b * M_ + i];
#pragma unroll
      for (int s = 1; s < SLICES_; ++s)
        m2 = fminf(m2, md2p[(size_t)s * BM_ + (size_t)b * M_ + i]);
      lc += m2;
    }
    sA[tid] = lc; __syncthreads();
    for (int s = 512; s > 0; s >>= 1) {
      if (tid < s) sA[tid] += sA[tid + s];
      __syncthreads();
    }
    if (tid == 0) total += loss1 + sA[0] / (float)M_;
    __syncthreads();
  }
  if (tid == 0) out[0] = total / (float)B_;
}

// ---------------------------------------------------------------------------
extern "C" void kernel_launch(void* const* d_in, const int* in_sizes, int n_in,
                              void* d_out, int out_size, void* d_ws, size_t ws_size,
                              hipStream_t stream) {
  const float* p1 = (const float*)d_in[0];   // (B,N,3)
  const float* p2 = (const float*)d_in[1];   // (B,M,3)
  const float* wt = (const float*)d_in[2];   // (B,N)
  float* out = (float*)d_out;

  // workspace carve-up (floats): ~2.4 MB total
  float* ws    = (float*)d_ws;
  float* a1aug = ws;                                  // B*N*4      = 131072
  float* a2aug = a1aug + (size_t)BN_ * 4;             // B*M*4      =  65536
  float* sq2   = a2aug + (size_t)BM_ * 4;             // B*M        =  16384
  float* md1p  = sq2 + (size_t)BM_;                   // SLICES*B*N = 262144
  float* md2p  = md1p + (size_t)SLICES_ * BN_;        // SLICES*B*M = 131072

  chamfer_prep<<<(BN_ + 255) / 256, 256, 0, stream>>>(p1, p2, a1aug, a2aug, sq2);

  // rowmin: B*(N/32)*8 = 8192 waves -> 1024 blocks of 8 waves
  chamfer_rowmin<<<(B_ * (N_ / 32) * SLICES_) / 8, 256, 0, stream>>>(
      a1aug, a2aug, sq2, md1p);

  // colmin: B*(M/32)*8 = 4096 waves -> 512 blocks of 8 waves
  chamfer_colmin<<<(B_ * (M_ / 32) * SLICES_) / 8, 256, 0, stream>>>(
      a1aug, a2aug, sq2, md2p);

  chamfer_final<<<1, 1024, 0, stream>>>(wt, md1p, md2p, out);
}